// LEGFamily_29798483100298
// MI455X (gfx1250) — compile-verified
//
#include <hip/hip_runtime.h>

typedef __attribute__((ext_vector_type(2))) float v2f;
typedef __attribute__((ext_vector_type(8))) float v8f;

#define WPB   8      // waves per block (256 threads)
#define ITERS 32     // pair-iterations per wave (64 pairs/wave)
#define THETA 0.25f  // scaling threshold for order-6 Taylor

// lane l <-> lane l^16 half-swap, pure VALU (v_permlanex16_b32).
__device__ __forceinline__ float swap_half(float x) {
  unsigned u = __builtin_bit_cast(unsigned, x);
  unsigned r = __builtin_amdgcn_permlanex16(u, u, 0x76543210u, 0xFEDCBA98u,
                                            false, false);
  return __builtin_bit_cast(float, r);
}

// broadcast lane ((l & ~7) | k) within each 8-lane group, pure VALU
// (v_permlane16_b32; selects are wave-uniform SGPR values).
__device__ __forceinline__ float bcast8(float x, int k) {
  unsigned u = __builtin_bit_cast(unsigned, x);
  unsigned sel_lo = (unsigned)k * 0x11111111u;        // lanes 0-7  -> k
  unsigned sel_hi = (unsigned)(k + 8) * 0x11111111u;  // lanes 8-15 -> 8+k
  unsigned r = __builtin_amdgcn_permlane16(u, u, sel_lo, sel_hi, false, false);
  return __builtin_bit_cast(float, r);
}

// Reinterpret a C/D-layout 16x16 f32 tile as a WMMA operand:
//  - used as A operand -> represents X^T
//  - used as B operand -> represents X
// (identical lane/VGPR mapping for both, per CDNA5 ISA layout tables)
__device__ __forceinline__ v8f conv_tile(v8f x) {
  const int lane = threadIdx.x & 31;
  const bool lo = lane < 16;
  float s0 = swap_half(x[0]);
  float s1 = swap_half(x[1]);
  float s2 = swap_half(x[2]);
  float s3 = swap_half(x[3]);
  float s4 = swap_half(x[4]);
  float s5 = swap_half(x[5]);
  float s6 = swap_half(x[6]);
  float s7 = swap_half(x[7]);
  v8f o;
  o[0] = lo ? x[0] : s2;
  o[4] = lo ? s0   : x[2];
  o[1] = lo ? x[1] : s3;
  o[5] = lo ? s1   : x[3];
  o[2] = lo ? x[4] : s6;
  o[6] = lo ? s4   : x[6];
  o[3] = lo ? x[5] : s7;
  o[7] = lo ? s5   : x[7];
  return o;
}

// Full 16x16x16 f32 product from 4 accumulating 16x16x4 WMMA slices.
// Computes P^T * Q when fed conv_tile(P), conv_tile(Q), plus acc.
__device__ __forceinline__ v8f wmma4(v8f A, v8f B, v8f acc) {
#pragma unroll
  for (int g = 0; g < 4; ++g) {
    v2f a, b;
    a[0] = A[2 * g]; a[1] = A[2 * g + 1];
    b[0] = B[2 * g]; b[1] = B[2 * g + 1];
    acc = __builtin_amdgcn_wmma_f32_16x16x4_f32(false, a, false, b,
                                                (short)0, acc, false, false);
  }
  return acc;
}

__global__ __launch_bounds__(WPB * 32)
void leg_kernel(const float* __restrict__ inp, const float* __restrict__ Np,
                const float* __restrict__ Rp, float* __restrict__ outDiag,
                float* __restrict__ outOff, int P) {
  __shared__ float Gs[65];                 // G (8x8) + ||G||_1
  __shared__ float augS[WPB][2][16][32];   // per-wave augmented systems
  __shared__ float xcS[WPB][2][128];       // per-wave X1/X2 diag blocks
  __shared__ float dcS[WPB][2][128];       // per-wave Dc1/Dc2 diag blocks

  const int tid = threadIdx.x;
  const int wid = tid >> 5;
  const int lane = tid & 31;

  // ---- build G = N N^T + R - R^T + 1e-5 I (redundantly per block) ----
  if (tid < 64) {
    int r = tid >> 3, c = tid & 7;
    float acc = (r == c) ? 1e-5f : 0.0f;
    int m = r < c ? r : c;
    for (int j = 0; j <= m; ++j)
      acc += Np[r * (r + 1) / 2 + j] * Np[c * (c + 1) / 2 + j];
    if (c < r) acc += Rp[r * (r - 1) / 2 + c];
    if (r < c) acc -= Rp[c * (c - 1) / 2 + r];
    Gs[tid] = acc;
  }
  __syncthreads();
  if (tid == 0) {
    float g = 0.f;
    for (int c = 0; c < 8; ++c) {
      float s = 0.f;
      for (int r = 0; r < 8; ++r) s += fabsf(Gs[r * 8 + c]);
      g = fmaxf(g, s);
    }
    Gs[64] = g;
  }
  __syncthreads();

  const int h = lane >> 4;              // M-block of this lane's elements
  const int nl = lane & 15;             // N (column) index
  const int ci = lane & 7;              // column within 8x8 block
  const bool inblk = (h == (nl >> 3));  // lane holds diag-block data
  const float gnorm = Gs[64];

  // Constant tiles (block-diagonal, both problems share G): C/D layout.
  v8f Gt, GTt, It;
#pragma unroll
  for (int k = 0; k < 8; ++k) {
    Gt[k]  = inblk ? Gs[k * 8 + ci] : 0.f;
    GTt[k] = inblk ? Gs[ci * 8 + k] : 0.f;
    It[k]  = (lane == k || lane == 24 + k) ? 1.f : 0.f;
  }

  const int gid = blockIdx.x * WPB + wid;
  const int base = gid * (2 * ITERS);
  const int lastPair = P - 1;
  if (base > lastPair) return;

  float carry0 = 0.f, carry1 = 0.f;  // Dc1[j0-1] flat (2 elems/lane)
  float (*aug)[16][32] = augS[wid];
  float* xc = &xcS[wid][0][0];
  float* dc = &dcS[wid][0][0];

  const int itStart = (base == 0) ? 0 : -1;  // pre-iteration fills carry
  for (int it = itStart; it < ITERS; ++it) {
    const int j0 = base + 2 * it;
    if (j0 > lastPair) break;
    const bool emit = (it >= 0);
    const int j1 = j0 + 1;
    const bool v1 = (j1 <= lastPair);

    float t0 = inp[j0];
    float t1 = inp[j0 + 1];
    float t2 = inp[v1 ? (j1 + 1) : (j0 + 1)];
    float d0 = t1 - t0;
    float d1 = v1 ? (t2 - t1) : 1.0f;

    // ---- scaling: ||A||_1 = 0.5*d*||G||_1, pick s so ||B|| <= THETA ----
    float an0 = 0.5f * d0 * gnorm;
    float an1 = 0.5f * d1 * gnorm;
    int s0 = 0, s1 = 0;
    if (an0 > THETA) { int v = (int)ceilf(log2f(an0 / THETA)); s0 = v > 30 ? 30 : v; }
    if (an1 > THETA) { int v = (int)ceilf(log2f(an1 / THETA)); s1 = v > 30 ? 30 : v; }
    const int smax = s0 > s1 ? s0 : s1;  // wave-uniform
    float sc0 = -ldexpf(0.5f * d0, -s0);
    float sc1 = -ldexpf(0.5f * d1, -s1);
    float scl = (lane < 16) ? sc0 : sc1;

    v8f Bt, BTt;
#pragma unroll
    for (int k = 0; k < 8; ++k) { Bt[k] = scl * Gt[k]; BTt[k] = scl * GTt[k]; }

    // ---- order-6 Taylor via Horner (WMMA): E=expm(B), ET=expm(B)^T ----
    const float Cc[7] = {1.f, 1.f, 0.5f, 1.f / 6.f, 1.f / 24.f, 1.f / 120.f, 1.f / 720.f};
    v8f E, ET;
#pragma unroll
    for (int k = 0; k < 8; ++k) {
      E[k]  = Cc[6] * Bt[k]  + Cc[5] * It[k];
      ET[k] = Cc[6] * BTt[k] + Cc[5] * It[k];
    }
    v8f cLE = conv_tile(BTt);  // (B^T)^T = B as left operand
    v8f cLT = conv_tile(Bt);   // B^T as left operand
#pragma unroll
    for (int kk = 4; kk >= 0; --kk) {
      v8f accE, accT;
#pragma unroll
      for (int k = 0; k < 8; ++k) { accE[k] = Cc[kk] * It[k]; accT[k] = Cc[kk] * It[k]; }
      E  = wmma4(cLE, conv_tile(E), accE);   // E  <- B*E    + c_k I
      ET = wmma4(cLT, conv_tile(ET), accT);  // ET <- B^T*ET + c_k I
    }

    // ---- squaring rounds (per-problem predicated) ----
    const int sp = (lane < 16) ? s0 : s1;
    for (int r_ = 0; r_ < smax; ++r_) {
      v8f ce = conv_tile(E);
      v8f ct = conv_tile(ET);
      v8f zz = {};
      v8f E2  = wmma4(ct, ce, zz);  // (ET)^T * E = E*E
      v8f ET2 = wmma4(ce, ct, zz);  // (E)^T * ET = ET*ET
      bool doit = r_ < sp;
#pragma unroll
      for (int k = 0; k < 8; ++k) {
        E[k]  = doit ? E2[k]  : E[k];
        ET[k] = doit ? ET2[k] : ET[k];
      }
    }

    // ---- Q = I - E^T E ;  P2 = I - E E^T ----
    v8f cE = conv_tile(E);
    v8f cT = conv_tile(ET);
    v8f z = {};
    v8f M1 = wmma4(cE, cE, z);  // E^T E
    v8f M2 = wmma4(cT, cT, z);  // E E^T
    v8f Q, P2;
#pragma unroll
    for (int k = 0; k < 8; ++k) { Q[k] = It[k] - M1[k]; P2[k] = It[k] - M2[k]; }

    // ---- two solves via LDS Gauss-Jordan (32 lanes = 2 sys x 2 prob x 8 rows)
#pragma unroll
    for (int k = 0; k < 8; ++k) {
      aug[0][k + 8 * h][nl]      = Q[k];
      aug[0][k + 8 * h][16 + nl] = ET[k];  // RHS: E^T
      aug[1][k + 8 * h][nl]      = P2[k];
      aug[1][k + 8 * h][16 + nl] = E[k];   // RHS: E
    }
    asm volatile("s_wait_dscnt 0" ::: "memory");
    const int sys = lane >> 4;
    const int pp = (lane >> 3) & 1;
    const int row = lane & 7;
    float a[16];
#pragma unroll
    for (int j = 0; j < 8; ++j) a[j] = aug[sys][8 * pp + row][8 * pp + j];
#pragma unroll
    for (int j = 0; j < 8; ++j) a[8 + j] = aug[sys][8 * pp + row][16 + 8 * pp + j];
#pragma unroll
    for (int k = 0; k < 8; ++k) {
      float pk[16];
#pragma unroll
      for (int j = 0; j < 16; ++j) pk[j] = bcast8(a[j], k);  // pivot row bcast (VALU)
      float inv = 1.0f / pk[k];
      if (row == k) {
#pragma unroll
        for (int j = 0; j < 16; ++j) a[j] *= inv;
      } else {
        float f = a[k] * inv;
#pragma unroll
        for (int j = 0; j < 16; ++j) a[j] -= f * pk[j];
      }
    }
#pragma unroll
    for (int j = 0; j < 8; ++j) xc[sys * 128 + pp * 64 + row * 8 + j] = a[8 + j];
    asm volatile("s_wait_dscnt 0" ::: "memory");

    // ---- X tiles back into C/D layout; Dc1 = E*X1, Dc2 = E^T*X2 ----
    v8f X1, X2;
#pragma unroll
    for (int k = 0; k < 8; ++k) {
      float u0 = xc[0 * 128 + h * 64 + k * 8 + ci];
      float u1 = xc[1 * 128 + h * 64 + k * 8 + ci];
      X1[k] = inblk ? u0 : 0.f;
      X2[k] = inblk ? u1 : 0.f;
    }
    v8f Dc1 = wmma4(cT, conv_tile(X1), z);
    v8f Dc2 = wmma4(cE, conv_tile(X2), z);

    // ---- stage diag-block data for coalesced output + carry ----
    if (inblk) {
#pragma unroll
      for (int k = 0; k < 8; ++k) {
        dc[0 * 128 + h * 64 + k * 8 + ci] = Dc1[k];
        dc[1 * 128 + h * 64 + k * 8 + ci] = Dc2[k];
      }
    }
    asm volatile("s_wait_dscnt 0" ::: "memory");

    if (emit) {
#pragma unroll
      for (int i2 = 0; i2 < 2; ++i2) {
        int e = lane + 32 * i2;
        float iv = ((e % 9) == 0) ? 1.f : 0.f;
        outOff[(size_t)j0 * 64 + e] = -xc[128 + e];            // -X2 (p0)
        float cv = (i2 == 0) ? carry0 : carry1;
        outDiag[(size_t)j0 * 64 + e] = iv + cv + dc[128 + e];  // I+Dc1[j0-1]+Dc2[j0]
      }
      if (v1) {
#pragma unroll
        for (int i2 = 0; i2 < 2; ++i2) {
          int e = lane + 32 * i2;
          float iv = ((e % 9) == 0) ? 1.f : 0.f;
          outOff[(size_t)j1 * 64 + e] = -xc[128 + 64 + e];          // -X2 (p1)
          outDiag[(size_t)j1 * 64 + e] = iv + dc[e] + dc[192 + e];  // I+Dc1[j0]+Dc2[j1]
        }
      } else {
        // j1 == P == T-1: final diag block = I + Dc1[T-2]
#pragma unroll
        for (int i2 = 0; i2 < 2; ++i2) {
          int e = lane + 32 * i2;
          float iv = ((e % 9) == 0) ? 1.f : 0.f;
          outDiag[(size_t)j1 * 64 + e] = iv + dc[e];
        }
      }
    }
    carry0 = dc[64 + lane];  // Dc1 (p1) becomes next carry
    carry1 = dc[64 + lane + 32];
  }
}

extern "C" void kernel_launch(void* const* d_in, const int* in_sizes, int n_in,
                              void* d_out, int out_size, void* d_ws, size_t ws_size,
                              hipStream_t stream) {
  const float* inp = (const float*)d_in[0];
  const float* Np = (const float*)d_in[1];
  const float* Rp = (const float*)d_in[2];
  const int T = in_sizes[0];
  const int P = T - 1;  // number of diffs / pairs
  float* outDiag = (float*)d_out;
  float* outOff = outDiag + (size_t)T * 64;

  const int pairsPerWave = 2 * ITERS;
  const int NW = (P + pairsPerWave - 1) / pairsPerWave;
  const int blocks = (NW + WPB - 1) / WPB;
  leg_kernel<<<blocks, WPB * 32, 0, stream>>>(inp, Np, Rp, outDiag, outOff, P);
}